// CRFClassifier_86681029968037
// MI455X (gfx1250) — compile-verified
//
#include <hip/hip_runtime.h>
#include <hip/hip_bf16.h>
#include <math.h>

typedef float v2f __attribute__((ext_vector_type(2)));
typedef float v8f __attribute__((ext_vector_type(8)));
typedef int   v4i __attribute__((ext_vector_type(4)));

#if defined(__has_builtin)
#if __has_builtin(__builtin_amdgcn_wmma_f32_16x16x4_f32)
#define CRF_HAVE_WMMA_F32X4 1
#endif
#if __has_builtin(__builtin_amdgcn_global_load_async_to_lds_b128)
#define CRF_HAVE_ASYNC 1
#endif
#if __has_builtin(__builtin_amdgcn_s_wait_asynccnt)
#define CRF_HAVE_WAIT_ASYNC 1
#endif
#endif
#ifndef CRF_HAVE_WMMA_F32X4
#define CRF_HAVE_WMMA_F32X4 0
#endif
#ifndef CRF_HAVE_ASYNC
#define CRF_HAVE_ASYNC 0
#endif
#ifndef CRF_HAVE_WAIT_ASYNC
#define CRF_HAVE_WAIT_ASYNC 0
#endif

constexpr int Bb = 64, Ss = 512, Hh = 1024, Ll = 33;
constexpr int NP   = 48;          // padded L (3 x 16 N-tiles)
constexpr int BK   = 64;          // K chunk staged in LDS
constexpr int LDA  = 68;          // LDS row stride (floats): 16B-aligned rows, conflict-free frags
constexpr int ROWS = 64;          // M rows per block (4 waves x 16)
constexpr int MTOT = Bb * Ss;     // 32768 rows
constexpr int NCH  = Hh / BK;     // 16 K-chunks

// ---- CDNA5 async global->LDS copy (ASYNCcnt path), sync fallback ----------
__device__ __forceinline__ void async_cp16(const float* g, float* l) {
#if CRF_HAVE_ASYNC
  __builtin_amdgcn_global_load_async_to_lds_b128(
      (__attribute__((address_space(1))) v4i*)g,
      (__attribute__((address_space(3))) v4i*)l, 0, 0);
#else
  *(float4*)l = *(const float4*)g;
#endif
}

__device__ __forceinline__ void wait_async0() {
#if CRF_HAVE_ASYNC
#if CRF_HAVE_WAIT_ASYNC
  __builtin_amdgcn_s_wait_asynccnt(0);
#else
  asm volatile("s_wait_asynccnt 0x0" ::: "memory");
#endif
#endif
}

// ---------------------------------------------------------------------------
// Kernel 1: emissions[b,s,l] = sum_h hs[b,s,h] * W[l,h] + bias[l]
// WMMA f32 16x16x4; double-buffered LDS with async global->LDS staging so the
// chunk-(c+1) DMA overlaps the chunk-c WMMAs (kernel is HBM-bound).
// ---------------------------------------------------------------------------
__global__ void __launch_bounds__(128)
emis_gemm_kernel(const float* __restrict__ hs, const float* __restrict__ W,
                 const float* __restrict__ bias, float* __restrict__ emis)
{
  __shared__ __align__(16) float As[2][ROWS * LDA];   // 2 x 17,408 B
  __shared__ __align__(16) float Ws[2][NP * LDA];     // 2 x 13,056 B

  const int tid  = threadIdx.x;
  const int wave = tid >> 5;
  const int lane = tid & 31;
  const int rows0 = blockIdx.x * ROWS;

  const int l15  = lane & 15;
  const int koff = (lane >> 4) * 2;   // high half-wave holds K+2 per ISA A/B layout

  // Zero-fill W pad rows (33..47) in both buffers once; DMA never touches them.
  for (int idx = tid; idx < (NP - Ll) * (BK / 4); idx += 128) {
    const int n = Ll + (idx >> 4), c4 = (idx & 15) * 4;
    const float4 z = make_float4(0.f, 0.f, 0.f, 0.f);
    *(float4*)(&Ws[0][n * LDA + c4]) = z;
    *(float4*)(&Ws[1][n * LDA + c4]) = z;
  }

  auto stage = [&](int buf, int k0) {
    // A chunk: 64 rows x 64 K, one b128 async per lane-iteration
    for (int idx = tid; idx < ROWS * (BK / 4); idx += 128) {
      const int r = idx >> 4, c4 = (idx & 15) * 4;
      async_cp16(hs + (size_t)(rows0 + r) * Hh + k0 + c4, &As[buf][r * LDA + c4]);
    }
    // W chunk: 33 live rows x 64 K
    for (int idx = tid; idx < Ll * (BK / 4); idx += 128) {
      const int n = idx >> 4, c4 = (idx & 15) * 4;
      async_cp16(W + (size_t)n * Hh + k0 + c4, &Ws[buf][n * LDA + c4]);
    }
  };

  v8f acc0 = {}, acc1 = {}, acc2 = {};

  stage(0, 0);
  wait_async0();
  __syncthreads();

  int buf = 0;
  for (int c = 0; c < NCH; ++c) {
    if (c + 1 < NCH) stage(buf ^ 1, (c + 1) * BK);   // DMA next chunk, no wait

    const float* Arow = &As[buf][(wave * 16 + l15) * LDA + koff];
    const float* Wn0  = &Ws[buf][(0 * 16 + l15) * LDA + koff];
    const float* Wn1  = &Ws[buf][(1 * 16 + l15) * LDA + koff];
    const float* Wn2  = &Ws[buf][(2 * 16 + l15) * LDA + koff];

#pragma unroll
    for (int kk = 0; kk < BK; kk += 4) {
      v2f a;  a.x  = Arow[kk]; a.y  = Arow[kk + 1];
      v2f b0; b0.x = Wn0[kk];  b0.y = Wn0[kk + 1];
      v2f b1; b1.x = Wn1[kk];  b1.y = Wn1[kk + 1];
      v2f b2; b2.x = Wn2[kk];  b2.y = Wn2[kk + 1];
#if CRF_HAVE_WMMA_F32X4
      acc0 = __builtin_amdgcn_wmma_f32_16x16x4_f32(false, a, false, b0, (short)0, acc0, false, false);
      acc1 = __builtin_amdgcn_wmma_f32_16x16x4_f32(false, a, false, b1, (short)0, acc1, false, false);
      acc2 = __builtin_amdgcn_wmma_f32_16x16x4_f32(false, a, false, b2, (short)0, acc2, false, false);
#else
      acc0[0] += a.x * b0.x + a.y * b0.y;
      acc1[0] += a.x * b1.x + a.y * b1.y;
      acc2[0] += a.x * b2.x + a.y * b2.y;
#endif
    }

    wait_async0();        // next chunk's DMA had the whole compute to land
    __syncthreads();      // all waves done reading buf before it is re-staged
    buf ^= 1;
  }

  // Epilogue: C/D layout -> VGPR v holds M = v (lanes 0-15) / v+8 (lanes 16-31)
  const int rbase = rows0 + wave * 16 + (lane >> 4) * 8;
#pragma unroll
  for (int v = 0; v < 8; ++v) {
    const int row = rbase + v;
    const int c0 = l15, c1 = 16 + l15, c2 = 32 + l15;
    emis[(size_t)row * NP + c0] = acc0[v] + bias[c0];
    emis[(size_t)row * NP + c1] = acc1[v] + bias[c1];
    if (c2 < Ll) emis[(size_t)row * NP + c2] = acc2[v] + bias[c2];
  }
}

// ---------------------------------------------------------------------------
// Kernel 2: per-batch CRF — gold score, forward logsumexp scan, Viterbi scan,
// LDS-resident backpointers (16.9 KB/batch), backtrace, masked pred write.
// One workgroup (64 threads = 2 waves) per batch; tags live in lanes 0..32.
// ---------------------------------------------------------------------------
__global__ void __launch_bounds__(64)
crf_scan_kernel(const float* __restrict__ emis, const int* __restrict__ mask,
                const int* __restrict__ labels, const float* __restrict__ startT,
                const float* __restrict__ endT, const float* __restrict__ trans,
                const int* __restrict__ padid,
                float* __restrict__ part, float* __restrict__ predOut)
{
  const int b = blockIdx.x, tid = threadIdx.x;

  __shared__ float T[Ll * Ll];           // 4.3 KB, for gold-score gathers
  __shared__ float alpha[Ll], vala[Ll];
  __shared__ float redf[64];
  __shared__ int   redi[64];
  __shared__ unsigned char bpS[Ss * Ll]; // 16.9 KB: backtrace stays in LDS
  __shared__ int   predS[Ss];
  __shared__ float scoreSh;

  for (int i = tid; i < Ll * Ll; i += 64) T[i] = trans[i];

  const float* emB = emis + (size_t)b * Ss * NP;
  const int*   mB  = mask   + b * Ss;
  const int*   lB  = labels + b * Ss;

  // Transition column for this thread's tag: loop-invariant over all 512 steps
  float tcol[Ll];
  if (tid < Ll) {
#pragma unroll
    for (int i = 0; i < Ll; ++i) tcol[i] = trans[i * Ll + tid];
    const float a0 = startT[tid] + emB[tid];
    alpha[tid] = a0; vala[tid] = a0;
  }
  __syncthreads();

  // ---- gold path score (parallel over t, deterministic tree reduce) ----
  float sacc = 0.f; int mcnt = 0;
  for (int t = tid; t < Ss; t += 64) {
    const int mt = mB[t]; mcnt += mt;
    if (t >= 1 && mt) {
      const int lt = lB[t], lp = lB[t - 1];
      sacc += emB[t * NP + lt] + T[lp * Ll + lt];
    }
  }
  redf[tid] = sacc; redi[tid] = mcnt;
  __syncthreads();
  for (int off = 32; off > 0; off >>= 1) {
    if (tid < off) { redf[tid] += redf[tid + off]; redi[tid] += redi[tid + off]; }
    __syncthreads();
  }
  if (tid == 0) {
    int lastIdx = redi[0] - 1; if (lastIdx < 0) lastIdx = 0;
    const int l0 = lB[0], lend = lB[lastIdx];
    scoreSh = redf[0] + startT[l0] + emB[l0] + endT[lend];
  }
  __syncthreads();

  // ---- fused forward (logsumexp, masked) + Viterbi (unmasked) scan ----
  for (int t = 1; t < Ss; ++t) {
    const int mt = mB[t];
    float na = 0.f, nv = 0.f; int am = 0;
    if (tid < Ll) {
      const float ej = emB[t * NP + tid];
      float s[Ll];
      float m1 = -INFINITY, m2 = -INFINITY;
#pragma unroll
      for (int i = 0; i < Ll; ++i) {
        const float tij = tcol[i];
        s[i] = alpha[i] + tij;
        m1 = fmaxf(m1, s[i]);
        const float vv = vala[i] + tij;
        if (vv > m2) { m2 = vv; am = i; }
      }
      float ssum = 0.f;
#pragma unroll
      for (int i = 0; i < Ll; ++i) ssum += __expf(s[i] - m1);
      na = m1 + __logf(ssum) + ej;
      nv = m2 + ej;
      bpS[t * Ll + tid] = (unsigned char)am;
    }
    __syncthreads();
    if (tid < Ll) { if (mt) alpha[tid] = na; vala[tid] = nv; }
    __syncthreads();
  }

  // ---- logZ, last tag, per-batch loss part, LDS backtrace ----
  if (tid == 0) {
    float m = -INFINITY;
    for (int j = 0; j < Ll; ++j) m = fmaxf(m, alpha[j] + endT[j]);
    float ssum = 0.f;
    for (int j = 0; j < Ll; ++j) ssum += __expf(alpha[j] + endT[j] - m);
    part[b] = scoreSh - (m + __logf(ssum));

    float bm = -INFINITY; int tag = 0;
    for (int j = 0; j < Ll; ++j) { const float v = vala[j] + endT[j]; if (v > bm) { bm = v; tag = j; } }
    predS[Ss - 1] = tag;
    for (int t = Ss - 1; t >= 1; --t) {          // ~40-cycle LDS hops, not HBM
      const int prev = bpS[t * Ll + tag];
      predS[t - 1] = prev;
      tag = prev;
    }
  }
  __syncthreads();

  const int pad = padid[0];
  for (int s = tid; s < Ss; s += 64) {
    int p = predS[s] * mB[s];
    if (p == 0) p = pad;
    predOut[(size_t)b * Ss + s] = (float)p;
  }
}

// ---------------------------------------------------------------------------
// Kernel 3: loss = -sum_b (score_b - logZ_b), deterministic single-block.
// ---------------------------------------------------------------------------
__global__ void __launch_bounds__(64)
loss_reduce_kernel(const float* __restrict__ part, float* __restrict__ out)
{
  __shared__ float r[64];
  r[threadIdx.x] = part[threadIdx.x];
  __syncthreads();
  for (int off = 32; off > 0; off >>= 1) {
    if (threadIdx.x < off) r[threadIdx.x] += r[threadIdx.x + off];
    __syncthreads();
  }
  if (threadIdx.x == 0) out[0] = -r[0];
}

// ---------------------------------------------------------------------------
extern "C" void kernel_launch(void* const* d_in, const int* in_sizes, int n_in,
                              void* d_out, int out_size, void* d_ws, size_t ws_size,
                              hipStream_t stream)
{
  const float* hs     = (const float*)d_in[0];   // (64,512,1024)
  const int*   mask   = (const int*)  d_in[1];   // (64,512)
  const int*   labels = (const int*)  d_in[2];   // (64,512)
  const float* W      = (const float*)d_in[3];   // (33,1024)
  const float* bias   = (const float*)d_in[4];   // (33)
  const float* startT = (const float*)d_in[5];   // (33)
  const float* endT   = (const float*)d_in[6];   // (33)
  const float* trans  = (const float*)d_in[7];   // (33,33)
  const int*   padid  = (const int*)  d_in[8];   // scalar

  // Workspace: emissions (padded N=48) | per-batch loss parts
  float* emis = (float*)d_ws;                    // 32768*48 f32 = 6.3 MB
  float* part = emis + (size_t)MTOT * NP;        // 64 f32

  float* out = (float*)d_out;   // out[0] = loss, out[1..] = pred (B*S) as float

  emis_gemm_kernel<<<MTOT / ROWS, 128, 0, stream>>>(hs, W, bias, emis);
  crf_scan_kernel<<<Bb, 64, 0, stream>>>(emis, mask, labels, startT, endT, trans,
                                         padid, part, out + 1);
  loss_reduce_kernel<<<1, 64, 0, stream>>>(part, out);
}